// InnerProduct_49185965474005
// MI455X (gfx1250) — compile-verified
//
#include <hip/hip_runtime.h>

typedef __attribute__((ext_vector_type(2))) float v2f;
typedef __attribute__((ext_vector_type(4))) float v4f;
typedef __attribute__((ext_vector_type(8))) float v8f;

#define FIELDS     39
#define DIM        64
#define NPAIR      741           // 39*38/2
#define ROWS_PAD   48            // 3 tiles of 16
#define LDS_STRIDE 68            // floats; 68*4B = 272B = 17*16B -> v4f aligned,
                                 // dword-bank stride 4 -> conflict-free frag reads
#define BLOCK      192           // 6 waves of 32

__global__ __launch_bounds__(BLOCK)
void InnerProduct_gram_wmma(const float* __restrict__ x,
                            float* __restrict__ out,
                            int batches) {
  __shared__ float xs[ROWS_PAD * LDS_STRIDE];

  const int b   = blockIdx.x;
  if (b >= batches) return;
  const int tid = threadIdx.x;
  const float* __restrict__ xb = x + (size_t)b * (FIELDS * DIM);

  // ---- Stage X[39][64] into LDS as 48 x 64 (rows 39..47 zero), stride 68 ----
  // 48 rows * 16 float4 = 768 v4f slots; 192 threads * 4 iterations.
  #pragma unroll
  for (int s = tid; s < ROWS_PAD * (DIM / 4); s += BLOCK) {
    const int row = s >> 4;        // 0..47
    const int cv  = s & 15;        // float4 column
    v4f val = {0.f, 0.f, 0.f, 0.f};
    if (row < FIELDS) {
      val = *(const v4f*)(xb + row * DIM + cv * 4);   // coalesced global_load_b128
    }
    *(v4f*)(&xs[row * LDS_STRIDE + cv * 4]) = val;
  }
  __syncthreads();

  // ---- Wave -> upper-triangular 16x16 tile of G = X * X^T ----
  const int wave = tid >> 5;                 // wave32
  const int lane = tid & 31;
  const int ti   = (wave < 3) ? 0 : ((wave < 5) ? 1 : 2);
  const int tj   = (wave < 3) ? wave : ((wave < 5) ? wave - 2 : 2);

  const int half = lane >> 4;                // 0: K={0,1}, 1: K={2,3}
  const int l16  = lane & 15;
  const int arow = ti * 16 + l16;            // A fragment: M = l16
  const int brow = tj * 16 + l16;            // B fragment: N = l16 (G = X*X^T)

  v8f acc = {};
  #pragma unroll
  for (int k0 = 0; k0 < DIM; k0 += 4) {
    const int ko = k0 + half * 2;
    v2f a = *(const v2f*)(&xs[arow * LDS_STRIDE + ko]);   // ds_load_b64
    v2f bb = *(const v2f*)(&xs[brow * LDS_STRIDE + ko]);  // ds_load_b64
    // D = A(16x4) * B(4x16) + C ; exact fp32 accumulate
    acc = __builtin_amdgcn_wmma_f32_16x16x4_f32(
        /*neg_a=*/false, a, /*neg_b=*/false, bb,
        /*c_mod=*/(short)0, acc, /*reuse_a=*/false, /*reuse_b=*/false);
  }

  // ---- Scatter strict upper triangle to packed pair index ----
  float* __restrict__ ob = out + (size_t)b * NPAIR;
  #pragma unroll
  for (int r = 0; r < 8; ++r) {
    const int m = ti * 16 + half * 8 + r;    // C/D layout: VGPR r, lane-half selects M/M+8
    const int c = tj * 16 + l16;
    if (m < FIELDS && c < FIELDS && m < c) {
      const int idx = m * (FIELDS - 1) - (m * (m - 1)) / 2 + (c - m - 1);
      ob[idx] = acc[r];
    }
  }
}

extern "C" void kernel_launch(void* const* d_in, const int* in_sizes, int n_in,
                              void* d_out, int out_size, void* d_ws, size_t ws_size,
                              hipStream_t stream) {
  const float* x = (const float*)d_in[0];
  float* out = (float*)d_out;
  const int batches = in_sizes[0] / (FIELDS * DIM);   // 4096
  InnerProduct_gram_wmma<<<batches, BLOCK, 0, stream>>>(x, out, batches);
}